// LC_24086176596634
// MI455X (gfx1250) — compile-verified
//
#include <hip/hip_runtime.h>

typedef __attribute__((ext_vector_type(2))) float v2f;
typedef __attribute__((ext_vector_type(8))) float v8f;

// ---------------------------------------------------------------------------
// f32 WMMA helper: D = A(16x4) * B(4x16) + C, full f32 precision
// ---------------------------------------------------------------------------
__device__ __forceinline__ v8f wmma4(v2f a, v2f b, v8f c) {
  return __builtin_amdgcn_wmma_f32_16x16x4_f32(false, a, false, b, (short)0, c,
                                               false, false);
}

// ---------------------------------------------------------------------------
// Tiled GEMM: C[M x N] = act(A[M x K] @ W[K x N] + bias)
// One wave computes a 64x64 tile as 4x4 WMMA sub-tiles: A fragments are
// reused across the 4 column tiles and B fragments across the 4 row tiles,
// so each K-step of 4 issues 16 v_wmma for only 4 b64 + 8 b32 loads.
// grid = (N/64, M/64); A, W, C row-major. RELU=1 adds bias + relu epilogue.
// ---------------------------------------------------------------------------
template <int RELU>
__global__ __launch_bounds__(32)
void gemm64_f32(const float* __restrict__ A, const float* __restrict__ W,
                const float* __restrict__ bias, float* __restrict__ C,
                int N, int K) {
  const int tn = blockIdx.x * 64;
  const int tm = blockIdx.y * 64;
  const int lane = threadIdx.x & 31;
  const int half = lane >> 4;   // selects K pair within the K=4 step
  const int idx  = lane & 15;   // row (A) / col (B)

  const float* Arow0 = A + (size_t)(tm + idx) * K;

  v8f acc[4][4];
#pragma unroll
  for (int rt = 0; rt < 4; ++rt)
#pragma unroll
    for (int ct = 0; ct < 4; ++ct) acc[rt][ct] = (v8f){};

  for (int k0 = 0; k0 < K; k0 += 4) {
    const int ka = k0 + half * 2;
    v2f a[4], b[4];
#pragma unroll
    for (int rt = 0; rt < 4; ++rt) {
      const float* Ar = Arow0 + (size_t)(rt * 16) * K;
      a[rt].x = Ar[ka];
      a[rt].y = Ar[ka + 1];
    }
#pragma unroll
    for (int ct = 0; ct < 4; ++ct) {
      b[ct].x = W[(size_t)ka * N + tn + ct * 16 + idx];
      b[ct].y = W[(size_t)(ka + 1) * N + tn + ct * 16 + idx];
    }
#pragma unroll
    for (int rt = 0; rt < 4; ++rt)
#pragma unroll
      for (int ct = 0; ct < 4; ++ct)
        acc[rt][ct] = wmma4(a[rt], b[ct], acc[rt][ct]);
  }

#pragma unroll
  for (int rt = 0; rt < 4; ++rt) {
#pragma unroll
    for (int ct = 0; ct < 4; ++ct) {
#pragma unroll
      for (int r = 0; r < 8; ++r) {
        const int m = tm + rt * 16 + r + 8 * half;
        const int n = tn + ct * 16 + idx;
        float v = acc[rt][ct][r];
        if (RELU) {
          v += bias[n];
          v = fmaxf(v, 0.0f);
        }
        C[(size_t)m * N + n] = v;
      }
    }
  }
}

// ---------------------------------------------------------------------------
// Fused per-(b,h) self-attention:  scores -> softmax (LDS) -> att = wce @ v
// Also exports wce[b,h,irow,:] (the only row needed by the gumbel adjacency).
// Block = 256 threads (8 waves).  LDS: 128x128 f32 scores = 64 KB.
// ---------------------------------------------------------------------------
__global__ __launch_bounds__(256)
void attn_fused(const float* __restrict__ q, const float* __restrict__ k,
                const float* __restrict__ v, float* __restrict__ att,
                float* __restrict__ wce_row, const int* __restrict__ id_robot) {
  __shared__ float s[128 * 128];
  const int bh = blockIdx.x;
  const int b = bh >> 3, h = bh & 7;
  const int tid = threadIdx.x;
  const int wave = tid >> 5;
  const int lane = tid & 31, half = lane >> 4, idx = lane & 15;
  const int irow = 128 - id_robot[0];

  const float* Q = q + (size_t)b * 128 * 256 + h * 32;
  const float* Kp = k + (size_t)b * 128 * 256 + h * 32;
  const float* Vp = v + (size_t)b * 128 * 256 + h * 32;
  const float scale = 0.17677669529663687f;  // 1/sqrt(32)

  // ---- scores = (q_h @ k_h^T) * scale ; wave handles 16-row band ----
  const int rm = wave * 16;
  for (int ct = 0; ct < 8; ++ct) {
    v8f acc = {};
    for (int k0 = 0; k0 < 32; k0 += 4) {
      const int ka = k0 + half * 2;
      v2f a, bb;
      a.x = Q[(rm + idx) * 256 + ka];
      a.y = Q[(rm + idx) * 256 + ka + 1];
      bb.x = Kp[(ct * 16 + idx) * 256 + ka];  // B(k,n) = K[n,k]  (k^T)
      bb.y = Kp[(ct * 16 + idx) * 256 + ka + 1];
      acc = wmma4(a, bb, acc);
    }
#pragma unroll
    for (int r = 0; r < 8; ++r)
      s[(rm + r + 8 * half) * 128 + ct * 16 + idx] = acc[r] * scale;
  }
  __syncthreads();

  // ---- row softmax in LDS (thread t owns row t) ----
  if (tid < 128) {
    float* row = s + tid * 128;
    float mx = -1e30f;
    for (int j = 0; j < 128; ++j) mx = fmaxf(mx, row[j]);
    float sum = 0.0f;
    for (int j = 0; j < 128; ++j) {
      const float e = expf(row[j] - mx);
      row[j] = e;
      sum += e;
    }
    const float inv = 1.0f / sum;
    for (int j = 0; j < 128; ++j) row[j] *= inv;
  }
  __syncthreads();

  if (tid < 128) wce_row[((size_t)b * 8 + h) * 128 + tid] = s[irow * 128 + tid];

  // ---- att_h = wce @ v_h  (128x32), wave handles row band x 2 col tiles ----
  for (int ct = 0; ct < 2; ++ct) {
    v8f acc = {};
    for (int k0 = 0; k0 < 128; k0 += 4) {
      const int ka = k0 + half * 2;
      v2f a, bb;
      a.x = s[(rm + idx) * 128 + ka];
      a.y = s[(rm + idx) * 128 + ka + 1];
      bb.x = Vp[ka * 256 + ct * 16 + idx];
      bb.y = Vp[(ka + 1) * 256 + ct * 16 + idx];
      acc = wmma4(a, bb, acc);
    }
#pragma unroll
    for (int r = 0; r < 8; ++r)
      att[((size_t)b * 128 + rm + r + 8 * half) * 256 + h * 32 + ct * 16 + idx] =
          acc[r];
  }
}

// ---------------------------------------------------------------------------
// Block reductions (128 threads)
// ---------------------------------------------------------------------------
__device__ __forceinline__ float brmax(float v, float* red) {
  const int t = threadIdx.x;
  red[t] = v;
  __syncthreads();
  for (int st = 64; st > 0; st >>= 1) {
    if (t < st) red[t] = fmaxf(red[t], red[t + st]);
    __syncthreads();
  }
  const float r = red[0];
  __syncthreads();
  return r;
}
__device__ __forceinline__ float brsum(float v, float* red) {
  const int t = threadIdx.x;
  red[t] = v;
  __syncthreads();
  for (int st = 64; st > 0; st >>= 1) {
    if (t < st) red[t] += red[t + st];
    __syncthreads();
  }
  const float r = red[0];
  __syncthreads();
  return r;
}

// ---------------------------------------------------------------------------
// Row-irow gumbel-sampled adjacency: adj_row[b,j] (only row needed downstream)
// ---------------------------------------------------------------------------
__global__ __launch_bounds__(128)
void adj_row_kernel(const float* __restrict__ wce_row,
                    const float* __restrict__ vis, const float* __restrict__ gum,
                    const float* __restrict__ w_lin,
                    const float* __restrict__ b_lin,
                    const int* __restrict__ id_robot,
                    float* __restrict__ adj_row) {
  __shared__ float red[128];
  const int b = blockIdx.x, j = threadIdx.x;
  const int irow = 128 - id_robot[0];
  const float wl = w_lin[0], bl = b_lin[0];
  const float visv = vis[((size_t)b * 128 + irow) * 128 + j];
  float acc = 0.0f;
  for (int h = 0; h < 8; ++h) {
    const float A = wce_row[((size_t)b * 8 + h) * 128 + j] * wl + bl;
    const float mx = brmax(A, red);
    const float e = expf(A - mx);
    const float se = brsum(e, red);
    float em = (e / se) * visv;
    const float ems = brsum(em, red);
    em = em / (ems + 1e-10f);
    const float u = gum[(((size_t)b * 8 + h) * 128 + irow) * 128 + j];
    const float g = -logf(-logf(u));
    const float t = logf(em + 1e-10f) + g;  // TAU = 1
    const float mx2 = brmax(t, red);
    const float e2 = expf(t - mx2);
    const float s2 = brsum(e2, red);
    acc += e2 / s2;
  }
  adj_row[b * 128 + j] = acc * visv;
}

// ---------------------------------------------------------------------------
// Single-row GAT: scores, softmax and head-averaged contraction for i = irow
// ---------------------------------------------------------------------------
__global__ __launch_bounds__(128)
void gat_row_kernel(const float* __restrict__ hbuf,
                    const float* __restrict__ a_src,
                    const float* __restrict__ a_tgt,
                    const float* __restrict__ adj_row,
                    const int* __restrict__ id_robot, float* __restrict__ out) {
  __shared__ float stgt[128 * 8];
  __shared__ float ssrc[8];
  __shared__ float aw[8 * 128];
  __shared__ float red[128];
  const int b = blockIdx.x, t = threadIdx.x;
  const int irow = 128 - id_robot[0];
  const float* Hb = hbuf + (size_t)b * 128 * 2048;

  // s_tgt[j,h] : 1024 dot products of length 256 (8 per thread)
  for (int r = 0; r < 8; ++r) {
    const int id = t + 128 * r;
    const int j = id >> 3, h = id & 7;
    float d = 0.0f;
    for (int f = 0; f < 256; ++f) d += Hb[j * 2048 + h * 256 + f] * a_tgt[h * 256 + f];
    stgt[j * 8 + h] = d;
  }
  if (t < 8) {
    float d = 0.0f;
    for (int f = 0; f < 256; ++f)
      d += Hb[irow * 2048 + t * 256 + f] * a_src[t * 256 + f];
    ssrc[t] = d;
  }
  __syncthreads();

  const float adjv = adj_row[b * 128 + t];
  for (int h = 0; h < 8; ++h) {
    float x = ssrc[h] + stgt[t * 8 + h];
    x = (x > 0.0f) ? x : 0.2f * x;  // leaky_relu(., 0.2)
    x += adjv;
    const float mx = brmax(x, red);
    const float e = expf(x - mx);
    const float s = brsum(e, red);
    aw[h * 128 + t] = e / s;
  }
  __syncthreads();

  // out[b, f] = (1/8) sum_h sum_j aw[h,j] * h[b,j,h,f]; thread does f=t, t+128
  float o0 = 0.0f, o1 = 0.0f;
  for (int j = 0; j < 128; ++j) {
#pragma unroll
    for (int h = 0; h < 8; ++h) {
      const float w = aw[h * 128 + j];
      o0 += w * Hb[j * 2048 + h * 256 + t];
      o1 += w * Hb[j * 2048 + h * 256 + t + 128];
    }
  }
  out[b * 256 + t] = o0 * 0.125f;
  out[b * 256 + t + 128] = o1 * 0.125f;
}

// ---------------------------------------------------------------------------
// Host launcher
// ---------------------------------------------------------------------------
extern "C" void kernel_launch(void* const* d_in, const int* in_sizes, int n_in,
                              void* d_out, int out_size, void* d_ws,
                              size_t ws_size, hipStream_t stream) {
  const float* x     = (const float*)d_in[0];   // [128,128,64]
  const float* vis   = (const float*)d_in[1];   // [128,128,128]
  const float* W_emb = (const float*)d_in[2];   // [64,256]
  const float* b_emb = (const float*)d_in[3];   // [256]
  const float* Wq    = (const float*)d_in[4];   // [256,256]
  const float* Wk    = (const float*)d_in[5];
  const float* Wv    = (const float*)d_in[6];
  const float* Wo    = (const float*)d_in[7];   // [256,256]
  const float* w_lin = (const float*)d_in[8];   // scalar
  const float* b_lin = (const float*)d_in[9];   // scalar
  const float* W_gat = (const float*)d_in[10];  // [256,2048]
  const float* a_src = (const float*)d_in[11];  // [8,256]
  const float* a_tgt = (const float*)d_in[12];  // [8,256]
  const float* gum   = (const float*)d_in[13];  // [128,8,128,128]
  const int*   idr   = (const int*)d_in[14];    // scalar
  float* out = (float*)d_out;                   // [128,256]

  // workspace layout (floats); buffers reused once dead:
  float* w    = (float*)d_ws;
  float* emb  = w;              // 4,194,304  (reused as `att`)
  float* qb   = w + 4194304;    // 4,194,304  (reused as `clustered`)
  float* kb   = w + 8388608;    // 4,194,304
  float* vb   = w + 12582912;   // 4,194,304
  float* hb   = w + 16777216;   // 33,554,432  h = clustered @ W_gat
  float* wrow = w + 50331648;   // 131,072     wce row irow per (b,h)
  float* arow = w + 50462720;   // 16,384      sampled adjacency row
  (void)in_sizes; (void)n_in; (void)out_size; (void)ws_size;

  const dim3 blk32(32);
  // emb = relu(x @ W_emb + b_emb)   [16384 x 64] @ [64 x 256]
  gemm64_f32<1><<<dim3(4, 256), blk32, 0, stream>>>(x, W_emb, b_emb, emb, 256, 64);
  // q,k,v = emb @ Wq/Wk/Wv          [16384 x 256] @ [256 x 256]
  gemm64_f32<0><<<dim3(4, 256), blk32, 0, stream>>>(emb, Wq, nullptr, qb, 256, 256);
  gemm64_f32<0><<<dim3(4, 256), blk32, 0, stream>>>(emb, Wk, nullptr, kb, 256, 256);
  gemm64_f32<0><<<dim3(4, 256), blk32, 0, stream>>>(emb, Wv, nullptr, vb, 256, 256);
  // fused attention per (b,h): att written into emb buffer (emb now dead)
  attn_fused<<<dim3(1024), dim3(256), 0, stream>>>(qb, kb, vb, emb, wrow, idr);
  // clustered = att @ Wo -> qb      [16384 x 256] @ [256 x 256]
  gemm64_f32<0><<<dim3(4, 256), blk32, 0, stream>>>(emb, Wo, nullptr, qb, 256, 256);
  // h = clustered @ W_gat -> hb     [16384 x 256] @ [256 x 2048]
  gemm64_f32<0><<<dim3(32, 256), blk32, 0, stream>>>(qb, W_gat, nullptr, hb, 2048, 256);
  // gumbel adjacency (row irow only) and final single-row GAT
  adj_row_kernel<<<dim3(128), dim3(128), 0, stream>>>(wrow, vis, gum, w_lin, b_lin, idr, arow);
  gat_row_kernel<<<dim3(128), dim3(128), 0, stream>>>(hb, a_src, a_tgt, arow, idr, out);
}